// GCN_8693013807111
// MI455X (gfx1250) — compile-verified
//
#include <hip/hip_runtime.h>
#include <math.h>

typedef __attribute__((ext_vector_type(2))) float v2f;
typedef __attribute__((ext_vector_type(4))) float v4f;
typedef __attribute__((ext_vector_type(8))) float v8f;

#define N_NODES 100000
#define N_EDGES 1600000
#define F_IN    256
#define HIDDEN  256
#define N_CLS   40
#define N_CLS_P 48   // N_CLS padded to a multiple of 16

// ---------------------------------------------------------------- utilities
__global__ void zero_f32(float* __restrict__ p, long long n) {
    long long i = (long long)blockIdx.x * blockDim.x + threadIdx.x;
    long long stride = (long long)gridDim.x * blockDim.x;
    for (; i < n; i += stride) p[i] = 0.0f;
}

__global__ void deg_scatter(const int* __restrict__ dst, const float* __restrict__ ew,
                            float* __restrict__ deg, int E) {
    int e = blockIdx.x * blockDim.x + threadIdx.x;
    if (e < E) atomicAdd(&deg[dst[e]], ew[e]);
}

// dis = deg^{-1/2} including the self-loop weight (+1)
__global__ void deg_finalize(const float* __restrict__ deg, float* __restrict__ dis, int n) {
    int i = blockIdx.x * blockDim.x + threadIdx.x;
    if (i < n) {
        float d = deg[i] + 1.0f;
        dis[i] = (d > 0.0f) ? rsqrtf(d) : 0.0f;
    }
}

__global__ void norm_kernel(const int* __restrict__ src, const int* __restrict__ dst,
                            const float* __restrict__ ew, const float* __restrict__ dis,
                            float* __restrict__ nrm, int E) {
    int e = blockIdx.x * blockDim.x + threadIdx.x;
    if (e < E) nrm[e] = dis[src[e]] * ew[e] * dis[dst[e]];
}

// zero-pad W2 [K x N_CLS] -> W2p [K x N_CLS_P], b2 -> b2p
__global__ void pad_w2(const float* __restrict__ W2, const float* __restrict__ b2,
                       float* __restrict__ W2p, float* __restrict__ b2p, int K) {
    int i = blockIdx.x * blockDim.x + threadIdx.x;
    int total = K * N_CLS_P;
    if (i < total) {
        int k = i / N_CLS_P, n = i % N_CLS_P;
        W2p[i] = (n < N_CLS) ? W2[k * N_CLS + n] : 0.0f;
    }
    if (i < N_CLS_P) b2p[i] = (i < N_CLS) ? b2[i] : 0.0f;
}

// ---------------------------------------------------------------- WMMA GEMM
// C[M,N] = A[M,K] @ B[K,N] + bias[N], fp32 via V_WMMA_F32_16X16X4_F32.
// One wave computes a 16 x (16*NSUB) tile. M must be a multiple of 16 and
// N an exact multiple of the column tiling -> no predication anywhere.
// Software-pipelined: iteration i issues loads for i+1 before running WMMAs.
template <int NSUB>
__global__ void gemm_wmma_f32(const float* __restrict__ A, const float* __restrict__ B,
                              const float* __restrict__ bias, float* __restrict__ C,
                              int M, int K, int N) {
    const int lane = threadIdx.x & 31;
    const int wave = threadIdx.x >> 5;
    const int half = lane >> 4;   // 0: lanes 0-15, 1: lanes 16-31
    const int l16  = lane & 15;

    const int mtile = blockIdx.x * (blockDim.x >> 5) + wave;
    const int m0 = mtile * 16;
    if (m0 >= M) return;          // uniform per wave -> EXEC stays all-1s below
    const int n0 = blockIdx.y * (16 * NSUB);

    v8f acc[NSUB];
#pragma unroll
    for (int t = 0; t < NSUB; ++t) {
        float bv = bias[n0 + t * 16 + l16];
#pragma unroll
        for (int r = 0; r < 8; ++r) acc[t][r] = bv;
    }

    // A fragment: lane holds row m0+l16, K pair {k0,k0+1}, k0 = kk + 2*half
    // B fragment: lane holds col n0+t*16+l16, K rows k0 and k0+1
    const float* ap = A + (long long)(m0 + l16) * K + 2 * half;
    const float* bp = B + (long long)(2 * half) * N + n0 + l16;
    const long long bstep = 4LL * N;

    // prologue: fragments for kk = 0
    v2f a_cur = *(const v2f*)ap;
    v2f b_cur[NSUB];
#pragma unroll
    for (int t = 0; t < NSUB; ++t) {
        b_cur[t].x = bp[t * 16];
        b_cur[t].y = bp[t * 16 + N];
    }

#pragma unroll 2
    for (int kk = 0; kk < K - 4; kk += 4) {
        ap += 4;
        bp += bstep;
        // issue next iteration's loads before consuming current fragments
        v2f a_nxt = *(const v2f*)ap;
        v2f b_nxt[NSUB];
#pragma unroll
        for (int t = 0; t < NSUB; ++t) {
            b_nxt[t].x = bp[t * 16];
            b_nxt[t].y = bp[t * 16 + N];
        }
#pragma unroll
        for (int t = 0; t < NSUB; ++t) {
            acc[t] = __builtin_amdgcn_wmma_f32_16x16x4_f32(
                false, a_cur, false, b_cur[t], (short)0, acc[t], false, false);
        }
        a_cur = a_nxt;
#pragma unroll
        for (int t = 0; t < NSUB; ++t) b_cur[t] = b_nxt[t];
    }
    // epilogue: last K-step
#pragma unroll
    for (int t = 0; t < NSUB; ++t) {
        acc[t] = __builtin_amdgcn_wmma_f32_16x16x4_f32(
            false, a_cur, false, b_cur[t], (short)0, acc[t], false, false);
    }

    float* crow = C + (long long)(m0 + 8 * half) * N + n0 + l16;
#pragma unroll
    for (int t = 0; t < NSUB; ++t) {
#pragma unroll
        for (int r = 0; r < 8; ++r) {
            crow[(long long)r * N + t * 16] = acc[t][r];
        }
    }
}

// ---------------------------------------------------------------- edge scatter
// One wave per edge; 256 features -> 8 per lane (two float4 chunks).
__global__ void scatter_edges_256(const int* __restrict__ src, const int* __restrict__ dst,
                                  const float* __restrict__ nrm, const float* __restrict__ h,
                                  float* __restrict__ agg, int E) {
    int wid  = (int)(((long long)blockIdx.x * blockDim.x + threadIdx.x) >> 5);
    int lane = threadIdx.x & 31;
    if (wid >= E) return;                 // uniform per wave
    int s = src[wid], d = dst[wid];
    float w = nrm[wid];
    const float* hs = h + (long long)s * 256;
    float* ad = agg + (long long)d * 256;
#pragma unroll
    for (int j = 0; j < 2; ++j) {
        int f = lane * 4 + j * 128;
        v4f v = *(const v4f*)(hs + f);
        atomicAdd(ad + f + 0, v.x * w);
        atomicAdd(ad + f + 1, v.y * w);
        atomicAdd(ad + f + 2, v.z * w);
        atomicAdd(ad + f + 3, v.w * w);
    }
}

// One wave per edge; h has leading dim N_CLS_P (48), agg has leading dim 40.
__global__ void scatter_edges_40(const int* __restrict__ src, const int* __restrict__ dst,
                                 const float* __restrict__ nrm, const float* __restrict__ h,
                                 float* __restrict__ agg, int E) {
    int wid  = (int)(((long long)blockIdx.x * blockDim.x + threadIdx.x) >> 5);
    int lane = threadIdx.x & 31;
    if (wid >= E) return;
    int s = src[wid], d = dst[wid];
    float w = nrm[wid];
    const float* hs = h + (long long)s * N_CLS_P;
    float* ad = agg + (long long)d * N_CLS;
    atomicAdd(ad + lane, hs[lane] * w);
    if (lane < 8) atomicAdd(ad + 32 + lane, hs[32 + lane] * w);
}

// agg = relu(agg + h * dis[node]^2), F = 256 (node = i >> 8)
__global__ void self_relu_256(const float* __restrict__ h, const float* __restrict__ dis,
                              float* __restrict__ agg, long long total) {
    long long i = (long long)blockIdx.x * blockDim.x + threadIdx.x;
    long long stride = (long long)gridDim.x * blockDim.x;
    for (; i < total; i += stride) {
        int node = (int)(i >> 8);
        float s = dis[node];
        float v = agg[i] + h[i] * s * s;
        agg[i] = fmaxf(v, 0.0f);
    }
}

// agg(ld 40) += h(ld 48) * dis[node]^2
__global__ void self_add_40(const float* __restrict__ h, const float* __restrict__ dis,
                            float* __restrict__ agg, long long total) {
    long long i = (long long)blockIdx.x * blockDim.x + threadIdx.x;
    long long stride = (long long)gridDim.x * blockDim.x;
    for (; i < total; i += stride) {
        int node = (int)(i / N_CLS);
        int f = (int)(i - (long long)node * N_CLS);
        float s = dis[node];
        agg[i] += h[(long long)node * N_CLS_P + f] * s * s;
    }
}

// in-place log_softmax over 40 classes; one wave per row
__global__ void log_softmax_40(float* __restrict__ out, int n) {
    int row  = (int)(((long long)blockIdx.x * blockDim.x + threadIdx.x) >> 5);
    int lane = threadIdx.x & 31;
    if (row >= n) return;
    float* p = out + (long long)row * N_CLS;
    float v0 = p[lane];
    float v1 = (lane < 8) ? p[lane + 32] : -3.4e38f;
    float m = fmaxf(v0, v1);
#pragma unroll
    for (int off = 16; off > 0; off >>= 1) m = fmaxf(m, __shfl_xor(m, off, 32));
    float s = expf(v0 - m) + ((lane < 8) ? expf(v1 - m) : 0.0f);
#pragma unroll
    for (int off = 16; off > 0; off >>= 1) s += __shfl_xor(s, off, 32);
    float lse = m + logf(s);
    p[lane] = v0 - lse;
    if (lane < 8) p[lane + 32] = v1 - lse;
}

// ---------------------------------------------------------------- launcher
extern "C" void kernel_launch(void* const* d_in, const int* in_sizes, int n_in,
                              void* d_out, int out_size, void* d_ws, size_t ws_size,
                              hipStream_t stream) {
    const float* x  = (const float*)d_in[0];
    const int*   ei = (const int*)d_in[1];     // [2, E]
    const float* ew = (const float*)d_in[2];
    const float* W1 = (const float*)d_in[3];
    const float* b1 = (const float*)d_in[4];
    const float* W2 = (const float*)d_in[5];
    const float* b2 = (const float*)d_in[6];
    float* out = (float*)d_out;

    const int N = N_NODES, E = N_EDGES;
    const int* src = ei;
    const int* dst = ei + E;

    // workspace carving (256B-aligned)
    char* ws = (char*)d_ws;
    size_t off = 0;
    auto carve = [&](size_t bytes) -> float* {
        float* p = (float*)(ws + off);
        off += (bytes + 255) & ~(size_t)255;
        return p;
    };
    float* deg  = carve((size_t)N * 4);
    float* dis  = carve((size_t)N * 4);
    float* b2p  = carve((size_t)N_CLS_P * 4);
    float* W2p  = carve((size_t)HIDDEN * N_CLS_P * 4);
    float* nrm  = carve((size_t)E * 4);
    float* h1   = carve((size_t)N * HIDDEN * 4);
    float* agg1 = carve((size_t)N * HIDDEN * 4);
    float* h2   = h1;  // h1 dead after self_relu_256; reuse as 100000 x 48

    const int T = 256;

    // 1) normalization coefficients + padded layer-2 weights
    zero_f32<<<(N + T - 1) / T, T, 0, stream>>>(deg, N);
    deg_scatter<<<(E + T - 1) / T, T, 0, stream>>>(dst, ew, deg, E);
    deg_finalize<<<(N + T - 1) / T, T, 0, stream>>>(deg, dis, N);
    norm_kernel<<<(E + T - 1) / T, T, 0, stream>>>(src, dst, ew, dis, nrm, E);
    pad_w2<<<(HIDDEN * N_CLS_P + T - 1) / T, T, 0, stream>>>(W2, b2, W2p, b2p, HIDDEN);

    // 2) layer 1: h1 = x @ W1 + b1  (100000 x 256 x 256), WMMA fp32
    {
        dim3 grid((N / 16 + 7) / 8, HIDDEN / 64);
        gemm_wmma_f32<4><<<grid, T, 0, stream>>>(x, W1, b1, h1, N, F_IN, HIDDEN);
    }

    // 3) edge aggregation + self-loop + ReLU
    long long tot1 = (long long)N * HIDDEN;
    zero_f32<<<100000, T, 0, stream>>>(agg1, tot1);
    scatter_edges_256<<<E / 8, T, 0, stream>>>(src, dst, nrm, h1, agg1, E);
    self_relu_256<<<100000, T, 0, stream>>>(h1, dis, agg1, tot1);

    // 4) layer 2: h2 = agg1 @ W2p + b2p  (100000 x 256 x 48, dense), WMMA fp32
    {
        dim3 grid((N / 16 + 7) / 8, 1);
        gemm_wmma_f32<3><<<grid, T, 0, stream>>>(agg1, W2p, b2p, h2, N, HIDDEN, N_CLS_P);
    }

    // 5) edge aggregation + self-loop into d_out (40-wide)
    long long tot2 = (long long)N * N_CLS;
    zero_f32<<<(int)((tot2 + T - 1) / T), T, 0, stream>>>(out, tot2);
    scatter_edges_40<<<E / 8, T, 0, stream>>>(src, dst, nrm, h2, out, E);
    self_add_40<<<(int)((tot2 + T - 1) / T), T, 0, stream>>>(h2, dis, out, tot2);

    // 6) log_softmax rows
    log_softmax_40<<<(N * 32 + T - 1) / T, T, 0, stream>>>(out, N);
}